// dca_layer_5875515261458
// MI455X (gfx1250) — compile-verified
//
#include <hip/hip_runtime.h>
#include <hip/hip_bf16.h>
#include <stdint.h>

// out[b,c,h,w] = x[b,c,h,w] * (1 + sigmoid(max_{h,w} x[b,c,:,:]))
// (the two channel shuffles in the reference cancel exactly)

#define HW        3136      // 56*56 floats per (b,c) slice = 12544 bytes
#define HW4       (HW / 4)  // 784 float4 chunks
#define BLOCK     256       // 8 wave32s
#define NWAVES    (BLOCK / 32)

typedef __attribute__((ext_vector_type(4))) unsigned int u32x4;
typedef __attribute__((ext_vector_type(4))) int          i32x4;
typedef __attribute__((ext_vector_type(8))) int          i32x8;

__global__ void __launch_bounds__(BLOCK)
dca_gate_kernel(const float* __restrict__ x, float* __restrict__ out, int n_bc) {
    __shared__ __align__(16) float tile[HW];
    __shared__ float wave_max[NWAVES];

    const int bc = blockIdx.x;
    if (bc >= n_bc) return;

    const float* __restrict__ src = x   + (size_t)bc * HW;
    float*       __restrict__ dst = out + (size_t)bc * HW;

#if __has_builtin(__builtin_amdgcn_tensor_load_to_lds) && \
    __has_builtin(__builtin_amdgcn_s_wait_tensorcnt)
    // ---- TDM path: one wave DMAs the whole 12.5KB slice into LDS ----
    if (threadIdx.x < 32) {  // wave 0 only (uniform branch -> other waves skip)
        const unsigned long long gaddr = (unsigned long long)(uintptr_t)src;
        const unsigned int lds_off = (unsigned int)(uintptr_t)(void*)tile; // low 32b = LDS offset

        // D# group0 (128b): count=1 | lds_addr | global_addr[56:0] | type=2
        u32x4 g0;
        g0.x = 1u;                                   // count = 1 valid descriptor
        g0.y = lds_off;                              // LDS byte address
        g0.z = (unsigned int)(gaddr & 0xFFFFFFFFull);            // global_addr[31:0]
        g0.w = (unsigned int)((gaddr >> 32) & 0x01FFFFFFull)     // global_addr[56:32]
             | (2u << 30);                                       // type = 2 ("image")

        // D# group1 (256b): data_size=4B, tensor_dim0=3136 (1 row), tile_dim0=3136
        i32x8 g1;
        g1[0] = (int)(2u << 16);                     // workgroup_mask=0, data_size=2 (4B)
        g1[1] = (int)(((unsigned)HW & 0xFFFFu) << 16);  // tensor_dim0[15:0] @ bit48
        g1[2] = (int)(((unsigned)HW >> 16) | (1u << 16)); // tensor_dim0[31:16] | tensor_dim1=1
        g1[3] = (int)(((unsigned)HW & 0xFFFFu) << 16);  // tile_dim0 = 3136 @ bit112
        g1[4] = 1;                                   // tile_dim1 = 1 (single row)
        g1[5] = (int)(unsigned)HW;                   // tensor_dim0_stride[31:0]
        g1[6] = (int)(((unsigned)HW & 0xFFFFu) << 16);  // dim1_stride[15:0] (unused, 1 row)
        g1[7] = 0;

        i32x4 g2 = {0, 0, 0, 0};                     // <=2D: groups 2/3 unused
        i32x4 g3 = {0, 0, 0, 0};

#if defined(__clang_major__) && (__clang_major__ >= 23)
        i32x8 g4 = {0, 0, 0, 0, 0, 0, 0, 0};
        __builtin_amdgcn_tensor_load_to_lds(g0, g1, g2, g3, g4, 0);
#else
        __builtin_amdgcn_tensor_load_to_lds(g0, g1, g2, g3, 0);
#endif
        __builtin_amdgcn_s_wait_tensorcnt(0);        // wave0's TENSORcnt -> 0
    }
#else
    // ---- Fallback: coalesced global -> LDS staging ----
    {
        const float4* __restrict__ s4 = (const float4*)src;
        float4* t4 = (float4*)tile;
        for (int i = threadIdx.x; i < HW4; i += BLOCK) t4[i] = s4[i];
    }
#endif
    __syncthreads();  // LDS tile visible to all 8 waves

    // ---- Per-slice max reduction out of LDS (ds_load_b128) ----
    float m = -3.402823466e38f;
    {
        const float4* t4 = (const float4*)tile;
        for (int i = threadIdx.x; i < HW4; i += BLOCK) {
            float4 v = t4[i];
            m = fmaxf(m, fmaxf(fmaxf(v.x, v.y), fmaxf(v.z, v.w)));
        }
    }
    // wave32 tree reduce
    #pragma unroll
    for (int off = 16; off > 0; off >>= 1)
        m = fmaxf(m, __shfl_xor(m, off, 32));
    if ((threadIdx.x & 31) == 0) wave_max[threadIdx.x >> 5] = m;
    __syncthreads();

    float gmax = wave_max[0];
    #pragma unroll
    for (int w = 1; w < NWAVES; ++w) gmax = fmaxf(gmax, wave_max[w]);

    // gate = 1 + sigmoid(gmax);  out = x * gate
    const float gate = 1.0f + 1.0f / (1.0f + __expf(-gmax));

    // ---- Scaled streamout: LDS -> global, b128 per lane ----
    {
        const float4* t4 = (const float4*)tile;
        float4* d4 = (float4*)dst;
        for (int i = threadIdx.x; i < HW4; i += BLOCK) {
            float4 v = t4[i];
            v.x *= gate; v.y *= gate; v.z *= gate; v.w *= gate;
            d4[i] = v;
        }
    }
}

extern "C" void kernel_launch(void* const* d_in, const int* in_sizes, int n_in,
                              void* d_out, int out_size, void* d_ws, size_t ws_size,
                              hipStream_t stream) {
    (void)n_in; (void)out_size; (void)d_ws; (void)ws_size;
    const float* x = (const float*)d_in[0];
    float* out = (float*)d_out;
    const int n_bc = in_sizes[0] / HW;   // 32 * 512 = 16384 slices
    dca_gate_kernel<<<n_bc, BLOCK, 0, stream>>>(x, out, n_bc);
}